// ESN_71330816852194
// MI455X (gfx1250) — compile-verified
//
#include <hip/hip_runtime.h>
#include <hip/hip_bf16.h>
#include <math.h>

// ---------------------------------------------------------------------------
// ESN: inp_proj = X @ W_in^T  (FP32 WMMA GEMM, chunked so proj stays L2-hot)
//      then s_t = tanh(u_t + d*s_{t-1}) scan (1 WGP, hw v_tanh_f32)
// ---------------------------------------------------------------------------

typedef float v2f __attribute__((ext_vector_type(2)));
typedef float v8f __attribute__((ext_vector_type(8)));

#define RES     1024      // reservoir size (N)
#define KDIM    128       // input size (K)
#define TM      32        // block tile M (rows of X)
#define TN      64        // block tile N (reservoir units)
#define APAD    132       // LDS row stride for A (32x128), bank-conflict free
#define BPAD    72        // LDS row stride for B^T (128x64): rows k and k+2
                          // map to disjoint 16-bank sets (72*2 % 64 == 16)

// Hardware tanh on the TRANS unit — single v_tanh_f32 on the 200000-deep
// dependent chain (software tanh would be ~15x slower there).
// NOTE: guard on __HIP_DEVICE_COMPILE__ — the HOST pass of hipcc does not
// declare amdgcn builtins (that's what tripped the previous #error probe).
// Device pass: builtin if declared, else inline asm; both emit v_tanh_f32.
__device__ __forceinline__ float fast_tanh(float x) {
#if defined(__HIP_DEVICE_COMPILE__)
#  if __has_builtin(__builtin_amdgcn_tanhf)
    return __builtin_amdgcn_tanhf(x);
#  elif __has_builtin(__builtin_amdgcn_tanh_f32)
    return __builtin_amdgcn_tanh_f32(x);
#  else
    float r;
    // v_nop satisfies the "1 independent op after TRANS before result use"
    // rule (ISA §7.4 case 3) since the compiler can't see through the asm.
    asm volatile("v_tanh_f32 %0, %1\n\tv_nop" : "=v"(r) : "v"(x));
    return r;
#  endif
#else
    return x;   // host pass stub: never executed (device-only call sites)
#endif
}

// ----------------------------- GEMM kernel --------------------------------
// grid: (ceil(Tc/32), 16), block: 256 threads (8 waves).
// wave w -> 16x16 tile at (mOff = (w&1)*16, nOff = (w>>1)*16).
__global__ __launch_bounds__(256)
void esn_gemm(const float* __restrict__ X, const float* __restrict__ Win,
              float* __restrict__ proj, int Tvalid)
{
    __shared__ float As[TM * APAD];      // X rows, row-major  [m][k]
    __shared__ float Bs[KDIM * BPAD];    // W_in^T,            [k][n]

    const int tid = threadIdx.x;
    const int m0  = blockIdx.x * TM;
    const int n0  = blockIdx.y * TN;

    // --- stage A: 32 rows x 128 K, float4 per thread, 4 passes ---
#pragma unroll
    for (int p = 0; p < 4; ++p) {
        int idx = p * 256 + tid;              // 0..1023
        int mr  = idx >> 5;                   // 0..31
        int k4  = (idx & 31) << 2;            // 0..124 step 4
        float4 v = make_float4(0.f, 0.f, 0.f, 0.f);
        if (m0 + mr < Tvalid)
            v = *(const float4*)(X + (size_t)(m0 + mr) * KDIM + k4);
        *(float4*)(&As[mr * APAD + k4]) = v;
    }
    // --- stage B transposed: 64 units x 128 K, coalesced read, scatter write
#pragma unroll
    for (int p = 0; p < 8; ++p) {
        int nr = p * 8 + (tid >> 5);          // 0..63
        int k4 = (tid & 31) << 2;             // 0..124 step 4
        float4 v = *(const float4*)(Win + (size_t)(n0 + nr) * KDIM + k4);
        Bs[(k4 + 0) * BPAD + nr] = v.x;
        Bs[(k4 + 1) * BPAD + nr] = v.y;
        Bs[(k4 + 2) * BPAD + nr] = v.z;
        Bs[(k4 + 3) * BPAD + nr] = v.w;
    }
    __syncthreads();

    const int wave = tid >> 5;
    const int lane = tid & 31;
    const int half = lane >> 4;               // 0: lanes 0-15, 1: lanes 16-31
    const int l16  = lane & 15;
    const int mOff = (wave & 1) * 16;
    const int nOff = (wave >> 1) * 16;

    // A frag: lane half 0 -> K = kb+{0,1}, half 1 -> K = kb+{2,3}, M = l16
    // B frag: same K split per half, N = l16
    const float* aBase = &As[(mOff + l16) * APAD + 2 * half];
    const float* bBase = &Bs[(2 * half) * BPAD + (nOff + l16)];

    v8f c = {};
#pragma unroll
    for (int kb = 0; kb < KDIM; kb += 4) {
        v2f a = *(const v2f*)(aBase + kb);
        v2f b;
        b.x = bBase[(kb + 0) * BPAD];
        b.y = bBase[(kb + 1) * BPAD];
        c = __builtin_amdgcn_wmma_f32_16x16x4_f32(
                /*neg_a=*/false, a, /*neg_b=*/false, b,
                /*c_mod=*/(short)0, c, /*reuse_a=*/false, /*reuse_b=*/false);
    }

    // C layout: VGPR r, lanes 0-15 -> (M=r, N=lane); 16-31 -> (M=8+r, N=l16)
    const int gn = n0 + nOff + l16;
    const int gmBase = m0 + mOff + 8 * half;
#pragma unroll
    for (int r = 0; r < 8; ++r) {
        int row = gmBase + r;
        if (row < Tvalid)
            proj[(size_t)row * RES + gn] = c[r];
    }
}

// ----------------------------- scan kernel --------------------------------
// One workgroup of 1024 threads = 32 waves on ONE WGP (8 waves/SIMD32):
// the dependent fma->tanh chains of 8 waves interleave per SIMD, and
// v_tanh_f32 (TRANS) co-executes with v_fma_f32 (VALU). Loads are chain-
// independent and nontemporal (single-use, L2-resident data).
__global__ __launch_bounds__(1024)
void esn_scan(const float* __restrict__ proj, const float* __restrict__ diag,
              float* __restrict__ state, float* __restrict__ out,
              int Tc, int isFirst, int isLast)
{
    const int i = threadIdx.x;
    const float d = diag[i];
    float s = isFirst ? 0.0f : state[i];

    const float* p = proj + i;
    int t = 0;
    for (; t + 8 <= Tc; t += 8) {
        float u0 = __builtin_nontemporal_load(p + 0 * RES);
        float u1 = __builtin_nontemporal_load(p + 1 * RES);
        float u2 = __builtin_nontemporal_load(p + 2 * RES);
        float u3 = __builtin_nontemporal_load(p + 3 * RES);
        float u4 = __builtin_nontemporal_load(p + 4 * RES);
        float u5 = __builtin_nontemporal_load(p + 5 * RES);
        float u6 = __builtin_nontemporal_load(p + 6 * RES);
        float u7 = __builtin_nontemporal_load(p + 7 * RES);
        s = fast_tanh(fmaf(d, s, u0));
        s = fast_tanh(fmaf(d, s, u1));
        s = fast_tanh(fmaf(d, s, u2));
        s = fast_tanh(fmaf(d, s, u3));
        s = fast_tanh(fmaf(d, s, u4));
        s = fast_tanh(fmaf(d, s, u5));
        s = fast_tanh(fmaf(d, s, u6));
        s = fast_tanh(fmaf(d, s, u7));
        p += 8 * RES;
    }
    for (; t < Tc; ++t) {
        s = fast_tanh(fmaf(d, s, __builtin_nontemporal_load(p)));
        p += RES;
    }

    state[i] = s;
    if (isLast) out[i] = s;
}

// ----------------------------- launcher -----------------------------------
extern "C" void kernel_launch(void* const* d_in, const int* in_sizes, int n_in,
                              void* d_out, int out_size, void* d_ws, size_t ws_size,
                              hipStream_t stream)
{
    const float* X    = (const float*)d_in[0];   // [T, 128]
    const float* Win  = (const float*)d_in[1];   // [1024, 128]
    const float* diag = (const float*)d_in[2];   // [1024]
    float* out = (float*)d_out;

    const long long T = (long long)in_sizes[0] / KDIM;

    // ws layout: [0,4K) carried state, [4K,...) proj chunk buffer
    float* state = (float*)d_ws;
    float* proj  = (float*)((char*)d_ws + 4096);

    // Chunk rows: cap at 8192 (32 MB) so GEMM stores stay resident in the
    // 192 MB L2 for the immediately-following scan read; adapt to ws_size.
    long long cap = ((long long)ws_size - 4096) / (RES * (long long)sizeof(float));
    long long Tc = cap < 8192 ? cap : 8192;
    Tc &= ~31LL;                      // multiple of TM
    if (Tc < 32) Tc = 32;

    for (long long base = 0; base < T; base += Tc) {
        long long rem = T - base;
        int cur = (int)(rem < Tc ? rem : Tc);
        dim3 grid((cur + TM - 1) / TM, RES / TN);
        esn_gemm<<<grid, 256, 0, stream>>>(X + base * KDIM, Win, proj, cur);
        esn_scan<<<1, 1024, 0, stream>>>(proj, diag, state, out, cur,
                                         base == 0 ? 1 : 0,
                                         (base + Tc >= T) ? 1 : 0);
    }
}